// TripletLoss_84490596647160
// MI455X (gfx1250) — compile-verified
//
#include <hip/hip_runtime.h>
#include <hip/hip_bf16.h>

typedef __attribute__((ext_vector_type(2))) float v2f;
typedef __attribute__((ext_vector_type(8))) float v8f;

// Problem shape from the reference setup_inputs()
constexpr int B_   = 16;
constexpr int M_   = 2048;
constexpr int D_   = 256;
constexpr int TILES = M_ / 16;         // 128 row-tiles per batch
constexpr float MARGIN = 5.0f;

// One wave32 per 16-row tile of the anchor.
// S_tile = sum_{m in tile} ( ||x_m - pos|| - ||x_m - neg|| )
// via V_WMMA_F32_16X16X4_F32 dot products and the Gram trick
// d^2 = |x|^2 + |y|^2 - 2 x.y, all in f32 (exact precision, BW-bound).
__global__ __launch_bounds__(256) void triplet_tile_kernel(
    const float* __restrict__ anchor,   // [B, M, D]
    const float* __restrict__ pos,      // [B, D]
    const float* __restrict__ neg,      // [B, D]
    float* __restrict__ partials) {     // [B * TILES]
  const int lane = threadIdx.x & 31;
  const int wave = blockIdx.x * (blockDim.x >> 5) + (threadIdx.x >> 5);
  const int b    = wave >> 7;            // wave / TILES
  const int tile = wave & (TILES - 1);

  const float* Pv = pos + b * D_;
  const float* Nv = neg + b * D_;

  // A-fragment layout (16x4 f32): lane L holds row m=L&15,
  // K-residues {kh, kh+1}, kh = (L>>4)*2.
  const int m  = lane & 15;
  const int kh = (lane >> 4) << 1;       // 0 or 2
  const float* rowbase =
      anchor + ((size_t)b * M_ + (size_t)tile * 16 + m) * D_;
  const float* rowp = rowbase + kh;

  // B-fragment (4x16 f32): lane L supplies column n=L&15, K-residues {kh,kh+1}.
  // Column 0 = pos, column 1 = neg.  Columns 2..15 of the product are never
  // read, so those lanes may load anything finite -> load pos, no branches.
  const float* bsrc = ((m == 1) ? Nv : Pv) + kh;

  // Burst-prefetch this wave's whole 16-row tile (16 KB) in 128B steps.
  #pragma unroll
  for (int j = 0; j < 8; ++j)
    __builtin_prefetch(rowbase + j * 32, 0, 3);

  v8f c = {0.f, 0.f, 0.f, 0.f, 0.f, 0.f, 0.f, 0.f};
  float sqx = 0.f;                       // half of |x_row|^2 (this lane's K-residues)

  #pragma unroll 8
  for (int i = 0; i < D_ / 4; ++i) {
    const int k0 = i * 4;
    v2f a  = *(const v2f*)(rowp + k0);
    v2f bf = *(const v2f*)(bsrc + k0);
    sqx = fmaf(a.x, a.x, sqx);
    sqx = fmaf(a.y, a.y, sqx);
    // 8 args: (neg_a, A, neg_b, B, c_mod, C, reuse_a, reuse_b)
    c = __builtin_amdgcn_wmma_f32_16x16x4_f32(
        false, a, false, bf, (short)0, c, false, false);
  }

  // Combine the two K-residue halves: lanes L and L^16 both get full |x_row|^2.
  sqx += __shfl_xor(sqx, 16, 32);

  // |pos|^2, |neg|^2 (tiny, wave-reduced; every lane ends with the full value).
  float sp = 0.f, sn = 0.f;
  #pragma unroll
  for (int j = 0; j < D_ / 32; ++j) {
    float p = Pv[lane + j * 32]; sp = fmaf(p, p, sp);
    float q = Nv[lane + j * 32]; sn = fmaf(q, q, sn);
  }
  #pragma unroll
  for (int off = 16; off >= 1; off >>= 1) {
    sp += __shfl_xor(sp, off, 32);
    sn += __shfl_xor(sn, off, 32);
  }

  // C/D layout: D[m][n], m<8 -> VGPR m, lane n ; m>=8 -> VGPR m-8, lane 16+n.
  float acc = 0.f;
  #pragma unroll
  for (int j = 0; j < 16; ++j) {
    float sq = __shfl(sqx, j, 32);                       // |x_row_j|^2
    float dp = __shfl(c[j & 7], (j < 8) ? 0 : 16, 32);   // x_j . pos
    float dn = __shfl(c[j & 7], (j < 8) ? 1 : 17, 32);   // x_j . neg
    float qp = fmaxf(fmaf(-2.f, dp, sq + sp), 0.f);      // safe_sqrt clamp
    float qn = fmaxf(fmaf(-2.f, dn, sq + sn), 0.f);
    acc += sqrtf(qp) - sqrtf(qn);
  }

  if (lane == 0) partials[b * TILES + tile] = acc;
}

// One wave: fixed-order (deterministic) reduction over tiles, then mean of
// relu((2/M)*S_b + margin) over batches.  (ed_x cancels exactly in the
// reference's distance_pos - distance_neg, so it is never computed.)
__global__ __launch_bounds__(32) void triplet_reduce_kernel(
    const float* __restrict__ partials, float* __restrict__ out) {
  const int lane = threadIdx.x;
  float loss = 0.f;
  if (lane < B_) {
    float s = 0.f;
    for (int t = 0; t < TILES; ++t) s += partials[lane * TILES + t];
    loss = fmaxf(fmaf(2.0f / (float)M_, s, MARGIN), 0.f);
  }
  #pragma unroll
  for (int off = 16; off >= 1; off >>= 1) loss += __shfl_xor(loss, off, 32);
  if (lane == 0) out[0] = loss / (float)B_;
}

extern "C" void kernel_launch(void* const* d_in, const int* in_sizes, int n_in,
                              void* d_out, int out_size, void* d_ws, size_t ws_size,
                              hipStream_t stream) {
  const float* anchor = (const float*)d_in[0];   // [16, 2048, 256] f32
  const float* pos    = (const float*)d_in[1];   // [16, 256] f32
  const float* neg    = (const float*)d_in[2];   // [16, 256] f32
  float* out          = (float*)d_out;           // scalar f32
  float* partials     = (float*)d_ws;            // B*TILES = 2048 floats (8 KB)

  // 16 batches * 128 tiles = 2048 waves; 8 waves (256 threads) per block.
  const int totalWaves = B_ * TILES;
  const int wavesPerBlock = 256 / 32;
  triplet_tile_kernel<<<totalWaves / wavesPerBlock, 256, 0, stream>>>(
      anchor, pos, neg, partials);
  triplet_reduce_kernel<<<1, 32, 0, stream>>>(partials, out);
}